// protein_gnn_17626545783635
// MI455X (gfx1250) — compile-verified
//
#include <hip/hip_runtime.h>

#define HD 128            // hidden width
#define POSF 16
#define EDGEF 32

typedef __attribute__((ext_vector_type(2))) float v2f;
typedef __attribute__((ext_vector_type(8))) float v8f;
typedef __attribute__((ext_vector_type(4))) unsigned int v4u;
typedef __attribute__((ext_vector_type(8))) int v8i;
typedef __attribute__((ext_vector_type(4))) int v4i;

// ---------------------------------------------------------------------------
// TDM: stage W [rows x cols] f32 (row-major, contiguous) into LDS at ldsOff.
// Descriptor per CDNA5 ISA:
//   group0: count=1 | lds_addr | global_addr[56:0] | type=2
//   group1: data_size=4B, pad every 128 DWORDs by 4 DWORDs (bank-conflict-free
//           LDS image: row stride = cols + 4*(cols/128) floats), dims/strides.
//   groups 2/3: zero (2-D tensor).
// This toolchain's builtin is the 6-arg form (extra int32x8 before cpol).
// Issued by one wave; completion via s_wait_tensorcnt then workgroup barrier.
// ---------------------------------------------------------------------------
__device__ __forceinline__ void tdm_stage_w(const float* W, unsigned ldsOff,
                                            int rows, int cols) {
    unsigned long long ga = (unsigned long long)(uintptr_t)W;
    v4u g0;
    g0[0] = 1u;                                        // count=1, user mode
    g0[1] = ldsOff;                                    // lds_addr (bytes)
    g0[2] = (unsigned)(ga & 0xFFFFFFFFu);              // global_addr lo
    g0[3] = (unsigned)((ga >> 32) & 0x01FFFFFFu)       // global_addr hi (bits 56:32)
            | (2u << 30);                              // type=2 ("image")
    v8i g1;
    unsigned dw0 = (2u << 16)                          // data_size = 4B
                 | (1u << 20)                          // pad_enable
                 | (6u << 22)                          // pad_interval: 128 DWORDs
                 | (3u << 25);                         // pad_amount: 4 DWORDs
    g1[0] = (int)dw0;                                  // workgroup_mask=0
    g1[1] = (int)((unsigned)(cols & 0xFFFF) << 16);    // tensor_dim0[15:0]
    g1[2] = (int)(((unsigned)cols >> 16) | ((unsigned)(rows & 0xFFFF) << 16)); // dim0 hi | dim1 lo
    g1[3] = (int)(((unsigned)rows >> 16) | ((unsigned)(cols & 0xFFFF) << 16)); // dim1 hi | tile_dim0
    g1[4] = (int)(rows & 0xFFFF);                      // tile_dim1 | tile_dim2=0
    g1[5] = (int)cols;                                 // tensor_dim0_stride lo32
    g1[6] = 0;                                         // stride0 hi | stride1 lo
    g1[7] = 0;                                         // stride1 hi
    v4i g2 = {0, 0, 0, 0};
    v4i g3 = {0, 0, 0, 0};
    v8i g4 = {0, 0, 0, 0, 0, 0, 0, 0};
    __builtin_amdgcn_tensor_load_to_lds(g0, g1, g2, g3, g4, 0);
}

// ---------------------------------------------------------------------------
// WMMA f32 GEMM:  C[M,128] = act( A[M,K] @ W[128,K]^T + bias ) (+ add)
// W staged to LDS by the TDM once per workgroup; A streamed from HBM.
// One wave computes a 16-row x 128-col tile with V_WMMA_F32_16X16X4_F32.
// CAT=true: A is the virtual concat [A | A2], each [M,128], K must be 256.
// ---------------------------------------------------------------------------
template <int K, bool RELU, bool HASADD, bool CAT>
__global__ __launch_bounds__(128) void gemm_wmma_f32(
    const float* __restrict__ A, const float* __restrict__ A2,
    const float* __restrict__ W, const float* __restrict__ bias,
    const float* __restrict__ addsrc, float* __restrict__ C, int M)
{
    extern __shared__ float smem[];                 // W image: 128 x RS floats
    constexpr int RS = K + 4 * (K / 128);           // padded LDS row stride

    if (threadIdx.x < 32) {                         // wave 0 issues the TDM op
        tdm_stage_w(W, (unsigned)(uintptr_t)smem, HD, K);
        __builtin_amdgcn_s_wait_tensorcnt(0);
    }
    __syncthreads();

    const int lane = threadIdx.x & 31;
    const int wave = threadIdx.x >> 5;
    const int m0   = (blockIdx.x * 4 + wave) * 16;
    if (m0 >= M) return;                            // wave-uniform
    const int r    = lane & 15;
    const int half = lane >> 4;                     // 0: K pair (k0,k0+1), 1: (k0+2,k0+3)

    v8f acc[8] = {};                                // 8 n-tiles of 16 cols -> N=128

    #pragma unroll 4
    for (int k0 = 0; k0 < K; k0 += 4) {
        const float* asrc;
        int kk;
        if (CAT && k0 >= HD) { asrc = A2; kk = k0 - HD; }
        else                 { asrc = A;  kk = k0; }
        // A fragment: 16x4, lane r holds row m0+r, this half's K pair
        const float* ap = asrc + (size_t)(m0 + r) * HD + kk + 2 * half;
        v2f a; a.x = ap[0]; a.y = ap[1];
        const int kIdx = k0 + ((K > HD && k0 >= HD) ? 4 : 0) + 2 * half; // skip mid-row pad
        #pragma unroll
        for (int nt = 0; nt < 8; nt++) {
            // B fragment from LDS: B[k][n] = W[n][k]; lane r holds col nt*16+r
            const float* wrow = smem + (nt * 16 + r) * RS + kIdx;
            v2f b; b.x = wrow[0]; b.y = wrow[1];
            acc[nt] = __builtin_amdgcn_wmma_f32_16x16x4_f32(
                false, a, false, b, (short)0, acc[nt], false, false);
        }
    }

    // Epilogue: acc[nt][v] is (m = m0 + v + 8*half, n = nt*16 + r)
    #pragma unroll
    for (int nt = 0; nt < 8; nt++) {
        const int n  = nt * 16 + r;
        const float bv = bias[n];
        #pragma unroll
        for (int v = 0; v < 8; v++) {
            const int m = m0 + v + 8 * half;
            float val = acc[nt][v] + bv;
            if (RELU)   val = fmaxf(val, 0.0f);
            if (HASADD) val += addsrc[(size_t)m * HD + n];
            C[(size_t)m * HD + n] = val;
        }
    }
}

// ---------------------------------------------------------------------------
// Fused edge pass for BOTH convs: per edge e
//   f0 = Wp @ pos[e] + bp   (128)     f1 = We @ ef[e] + be   (128)
//   agg0[dst] += f0 * x1[src];  agg1[dst] += f1 * x1[src];  cnt[dst] += 1
// One wave per edge; projection weights register-resident per lane-column.
// ---------------------------------------------------------------------------
__global__ __launch_bounds__(256) void edge_pass(
    const float* __restrict__ x1,
    const float* __restrict__ pos, const float* __restrict__ ef,
    const int* __restrict__ srcIdx, const int* __restrict__ dstIdx,
    const float* __restrict__ Wp, const float* __restrict__ bp,
    const float* __restrict__ We, const float* __restrict__ be,
    float* __restrict__ agg0, float* __restrict__ agg1,
    float* __restrict__ cnt, int E)
{
    const int lane = threadIdx.x & 31;

    // Per-lane weight columns: channel c = lane + 32*j
    float wp[4][POSF];
    float we[4][EDGEF];
    float b0[4], b1[4];
    #pragma unroll
    for (int j = 0; j < 4; j++) {
        const int c = lane + 32 * j;
        b0[j] = bp[c];
        b1[j] = be[c];
        #pragma unroll
        for (int k = 0; k < POSF; k++)  wp[j][k] = Wp[c * POSF + k];
        #pragma unroll
        for (int k = 0; k < EDGEF; k++) we[j][k] = We[c * EDGEF + k];
    }

    const int gw = (int)((blockIdx.x * blockDim.x + threadIdx.x) >> 5);
    const int nw = (int)((gridDim.x * blockDim.x) >> 5);

    for (int e = gw; e < E; e += nw) {
        const int s = srcIdx[e];
        const int d = dstIdx[e];
        const float p = pos[(size_t)e * POSF + (lane & (POSF - 1))];
        const float q = ef[(size_t)e * EDGEF + lane];
        if (lane == 0) unsafeAtomicAdd(&cnt[d], 1.0f);

        float f0[4], f1[4];
        #pragma unroll
        for (int j = 0; j < 4; j++) { f0[j] = b0[j]; f1[j] = b1[j]; }
        #pragma unroll
        for (int k = 0; k < POSF; k++) {
            const float pk = __shfl(p, k, 32);
            #pragma unroll
            for (int j = 0; j < 4; j++) f0[j] = fmaf(wp[j][k], pk, f0[j]);
        }
        #pragma unroll
        for (int k = 0; k < EDGEF; k++) {
            const float qk = __shfl(q, k, 32);
            #pragma unroll
            for (int j = 0; j < 4; j++) f1[j] = fmaf(we[j][k], qk, f1[j]);
        }
        #pragma unroll
        for (int j = 0; j < 4; j++) {
            const int c = lane + 32 * j;
            const float xv = x1[(size_t)s * HD + c];   // gather (L2-resident)
            unsafeAtomicAdd(&agg0[(size_t)d * HD + c], f0[j] * xv);
            unsafeAtomicAdd(&agg1[(size_t)d * HD + c], f1[j] * xv);
        }
    }
}

__global__ void zero_kernel(float* __restrict__ p, long long n) {
    long long i = (long long)blockIdx.x * blockDim.x + threadIdx.x;
    if (i < n) p[i] = 0.0f;
}

__global__ void mean_div(float* __restrict__ agg0, float* __restrict__ agg1,
                         const float* __restrict__ cnt, long long total) {
    long long i = (long long)blockIdx.x * blockDim.x + threadIdx.x;
    if (i >= total) return;
    const float inv = 1.0f / fmaxf(cnt[i >> 7], 1.0f);
    agg0[i] *= inv;
    agg1[i] *= inv;
}

// ---------------------------------------------------------------------------
extern "C" void kernel_launch(void* const* d_in, const int* in_sizes, int n_in,
                              void* d_out, int out_size, void* d_ws, size_t ws_size,
                              hipStream_t stream) {
    (void)n_in; (void)out_size; (void)ws_size;

    const int N = in_sizes[0] / HD;       // 50000
    const int E = in_sizes[1] / POSF;     // 600000

    const float* x    = (const float*)d_in[0];
    const float* pos  = (const float*)d_in[1];
    const float* efeat= (const float*)d_in[2];
    const int*   idx  = (const int*)d_in[3];
    const int*   srcI = idx;              // edge_index row 0 = source
    const int*   dstI = idx + E;          // edge_index row 1 = dest

    // params: {w,b} pairs in dict order
    const float* Wx1 = (const float*)d_in[4];  const float* bx1 = (const float*)d_in[5];
    const float* Wx2 = (const float*)d_in[6];  const float* bx2 = (const float*)d_in[7];
    const float* Wp1 = (const float*)d_in[8];  const float* bp1 = (const float*)d_in[9];
    const float* Wp2 = (const float*)d_in[10]; const float* bp2 = (const float*)d_in[11];
    const float* We1 = (const float*)d_in[12]; const float* be1 = (const float*)d_in[13];
    const float* We2 = (const float*)d_in[14]; const float* be2 = (const float*)d_in[15];
    const float* Wc0l= (const float*)d_in[16]; const float* bc0l= (const float*)d_in[17];
    const float* Wc0r= (const float*)d_in[18]; const float* bc0r= (const float*)d_in[19];
    const float* Wc1l= (const float*)d_in[20]; const float* bc1l= (const float*)d_in[21];
    const float* Wc1r= (const float*)d_in[22]; const float* bc1r= (const float*)d_in[23];
    const float* Wct0= (const float*)d_in[24]; const float* bct0= (const float*)d_in[25];
    const float* Wct1= (const float*)d_in[26]; const float* bct1= (const float*)d_in[27];
    const float* Wfin= (const float*)d_in[28]; const float* bfin= (const float*)d_in[29];

    float* out = (float*)d_out;

    const long long NB = (long long)N * HD;
    float* x1   = (float*)d_ws;
    float* x2   = x1 + NB;
    float* agg0 = x2 + NB;
    float* agg1 = agg0 + NB;
    float* cnt  = agg1 + NB;

    const dim3 gblk(128);
    const dim3 ggrd((N + 63) / 64);
    const unsigned shm128 = 128u * (HD + 4) * sizeof(float);        // 67,584 B
    const unsigned shm256 = 128u * (2 * HD + 8) * sizeof(float);    // 135,168 B

    // 0. zero agg0|agg1|cnt (contiguous)
    {
        long long total = 2 * NB + N;
        zero_kernel<<<dim3((unsigned)((total + 255) / 256)), dim3(256), 0, stream>>>(agg0, total);
    }
    // 1-2. x_lin_1 / x_lin_2 = relu(x @ W^T + b)
    gemm_wmma_f32<HD, true, false, false><<<ggrd, gblk, shm128, stream>>>(x, nullptr, Wx1, bx1, nullptr, x1, N);
    gemm_wmma_f32<HD, true, false, false><<<ggrd, gblk, shm128, stream>>>(x, nullptr, Wx2, bx2, nullptr, x2, N);
    // 3. fused edge projections + gather + scatter-add (both convs)
    edge_pass<<<dim3(1024), dim3(256), 0, stream>>>(x1, pos, efeat, srcI, dstI,
                                                    Wp1, bp1, We1, be1, agg0, agg1, cnt, E);
    // 4. mean aggregation
    mean_div<<<dim3((unsigned)((NB + 255) / 256)), dim3(256), 0, stream>>>(agg0, agg1, cnt, NB);
    // 5. conv0: h0 = lin_l(agg0) + lin_r(x1)    (in-place on agg0)
    gemm_wmma_f32<HD, false, false, false><<<ggrd, gblk, shm128, stream>>>(agg0, nullptr, Wc0l, bc0l, nullptr, agg0, N);
    gemm_wmma_f32<HD, false, true,  false><<<ggrd, gblk, shm128, stream>>>(x1,   nullptr, Wc0r, bc0r, agg0,   agg0, N);
    // 6. h0 = relu(lin_pos2(h0))
    gemm_wmma_f32<HD, true, false, false><<<ggrd, gblk, shm128, stream>>>(agg0, nullptr, Wp2, bp2, nullptr, agg0, N);
    // 7. conv1: h1 = lin_l(agg1) + lin_r(x1)
    gemm_wmma_f32<HD, false, false, false><<<ggrd, gblk, shm128, stream>>>(agg1, nullptr, Wc1l, bc1l, nullptr, agg1, N);
    gemm_wmma_f32<HD, false, true,  false><<<ggrd, gblk, shm128, stream>>>(x1,   nullptr, Wc1r, bc1r, agg1,   agg1, N);
    // 8. h1 = relu(lin_edge2(h1))
    gemm_wmma_f32<HD, true, false, false><<<ggrd, gblk, shm128, stream>>>(agg1, nullptr, We2, be2, nullptr, agg1, N);
    // 9. h = relu(cat(h0,h1) @ Wcat0^T + b)   (virtual concat, K=256)
    gemm_wmma_f32<2 * HD, true, false, true><<<ggrd, gblk, shm256, stream>>>(agg0, agg1, Wct0, bct0, nullptr, agg0, N);
    // 10. h = relu(lins_cat1(h)) + x_lin_2
    gemm_wmma_f32<HD, true, true, false><<<ggrd, gblk, shm128, stream>>>(agg0, nullptr, Wct1, bct1, x2, agg0, N);
    // 11. out = final(h)
    gemm_wmma_f32<HD, false, false, false><<<ggrd, gblk, shm128, stream>>>(agg0, nullptr, Wfin, bfin, nullptr, out, N);
}